// DeepseekV3Attention_86526411145670
// MI455X (gfx1250) — compile-verified
//
#include <hip/hip_runtime.h>
#include <hip/hip_bf16.h>

// ---------------------------------------------------------------------------
// DeepSeek-V3 MLA attention prefill for gfx1250 (MI455X).
// f32 matrix math via V_WMMA_F32_16X16X4_F32 (fast f32 path on CDNA5),
// data movement via GLOBAL_LOAD_ASYNC_TO_LDS_B128 + double-buffered LDS
// (ASYNCcnt pipeline), so DMA of stage s+1 overlaps WMMAs of stage s.
// ---------------------------------------------------------------------------

typedef float v2f __attribute__((ext_vector_type(2)));
typedef float v8f __attribute__((ext_vector_type(8)));

#define WMMA_F32(a, b, c) \
  __builtin_amdgcn_wmma_f32_16x16x4_f32(false, (a), false, (b), (short)0, (c), false, false)

constexpr int S_      = 2048;
constexpr int D_      = 2048;
constexpr int H_      = 16;
constexpr int KVH_    = 4;     // KV heads
constexpr int D_QK_   = 192;
constexpr int D_NOPE_ = 128;
constexpr int D_ROPE_ = 64;
constexpr int D_V_    = 128;
constexpr int QRANK_  = 1536;
constexpr int KVRANK_ = 512;

// Async DMA: global -> LDS, 16 bytes per lane. LDS offset is the low 32 bits
// of the generic pointer (LDS aperture keeps the wave-relative offset there).
__device__ __forceinline__ void async_ld_b128(const float* g, const float* lds_elem)
{
  uint32_t l = (uint32_t)(uintptr_t)lds_elem;
  asm volatile("global_load_async_to_lds_b128 %0, %1, off"
               :: "v"(l), "v"(g) : "memory");
}
__device__ __forceinline__ void wait_async0()
{
  asm volatile("s_wait_asynccnt 0x0" ::: "memory");
}

// ---------------------------------------------------------------------------
// f32 WMMA GEMM: C[M,N] = A[M,K] * B[K,N], row-major.
// Block = 128 threads (4 waves), block tile 64x64, K staged 32 wide through
// double-buffered LDS via async DMA. Wave w owns rows [bY*64+16w, +16).
// Requires M%64==0, N%64==0, K%32==0 (true at every call site).
// ---------------------------------------------------------------------------
__global__ __launch_bounds__(128)
void gemm_wmma_f32(const float* __restrict__ A, const float* __restrict__ B,
                   float* __restrict__ C, int K, int lda, int ldb, int ldc)
{
  __shared__ float sA[2][64 * 32];   // 8 KB per buffer
  __shared__ float sB[2][32 * 64];   // 8 KB per buffer

  const int tid   = threadIdx.x;
  const int lane  = tid & 31;
  const int wave  = tid >> 5;
  const int mrow  = lane & 15;       // M (A-frag) / N (B-frag) index in tile
  const int half  = lane >> 4;       // selects K pair {0,1} vs {2,3}
  const int khalf = half * 2;
  const int m0 = blockIdx.y * 64;
  const int n0 = blockIdx.x * 64;

  // One stage: A 64x32 (512 float4), B 32x64 (512 float4); 4+4 b128 per thread.
  auto issue_stage = [&](int k0, int buf) {
#pragma unroll
    for (int i = 0; i < 4; ++i) {
      int f  = tid + 128 * i;
      int ra = f >> 3, ca = (f & 7) << 2;
      async_ld_b128(A + (size_t)(m0 + ra) * lda + k0 + ca, &sA[buf][ra * 32 + ca]);
    }
#pragma unroll
    for (int i = 0; i < 4; ++i) {
      int f  = tid + 128 * i;
      int rb = f >> 4, cb = (f & 15) << 2;
      async_ld_b128(B + (size_t)(k0 + rb) * ldb + n0 + cb, &sB[buf][rb * 64 + cb]);
    }
  };

  v8f acc0 = {}, acc1 = {}, acc2 = {}, acc3 = {};

  issue_stage(0, 0);
  const int nstage = K >> 5;
  for (int st = 0; st < nstage; ++st) {
    wait_async0();        // this thread's stage-st DMA done
    __syncthreads();      // everyone's DMA done; prev-stage readers done
    if (st + 1 < nstage) issue_stage((st + 1) << 5, (st + 1) & 1);

    const float* Aw = &sA[st & 1][(wave * 16 + mrow) * 32 + khalf];
    const float* Bw = &sB[st & 1][mrow];
#pragma unroll
    for (int kk = 0; kk < 8; ++kk) {
      v2f a = *(const v2f*)(Aw + kk * 4);
      const float* b0 = Bw + (kk * 4 + khalf) * 64;
      const float* b1 = b0 + 64;
      v2f bb;
      bb.x = b0[0];  bb.y = b1[0];  acc0 = WMMA_F32(a, bb, acc0);
      bb.x = b0[16]; bb.y = b1[16]; acc1 = WMMA_F32(a, bb, acc1);
      bb.x = b0[32]; bb.y = b1[32]; acc2 = WMMA_F32(a, bb, acc2);
      bb.x = b0[48]; bb.y = b1[48]; acc3 = WMMA_F32(a, bb, acc3);
    }
    __syncthreads();      // reads of buf st&1 done before it is re-filled
  }

  // D layout: VGPR r holds row (m0 + 16*wave + r + half*8), col n0 + mrow.
  float* Crow = C + (size_t)(m0 + wave * 16 + half * 8) * ldc + n0 + mrow;
#pragma unroll
  for (int r = 0; r < 8; ++r) {
    Crow[(size_t)r * ldc + 0]  = acc0[r];
    Crow[(size_t)r * ldc + 16] = acc1[r];
    Crow[(size_t)r * ldc + 32] = acc2[r];
    Crow[(size_t)r * ldc + 48] = acc3[r];
  }
}

// ---------------------------------------------------------------------------
// Row-wise RMSNorm: out[row,:w] = in[row,:w] * rsqrt(mean(x^2)+eps) * weight
// ---------------------------------------------------------------------------
__global__ __launch_bounds__(256)
void rmsnorm_rows(const float* __restrict__ in, const float* __restrict__ w,
                  float* __restrict__ out, int width, int in_ld, int out_ld)
{
  __shared__ float red[256];
  const int row = blockIdx.x;
  const float* x = in + (size_t)row * in_ld;
  float ss = 0.f;
  for (int i = threadIdx.x; i < width; i += 256) {
    float v = x[i];
    ss += v * v;
  }
  red[threadIdx.x] = ss;
  __syncthreads();
  for (int s = 128; s > 0; s >>= 1) {
    if ((int)threadIdx.x < s) red[threadIdx.x] += red[threadIdx.x + s];
    __syncthreads();
  }
  const float scale = rsqrtf(red[0] / (float)width + 1e-6f);
  float* y = out + (size_t)row * out_ld;
  for (int i = threadIdx.x; i < width; i += 256) y[i] = x[i] * scale * w[i];
}

// ---------------------------------------------------------------------------
// RoPE on shared k_rot (ckv[:,512:576]) -> krot[S,64]; pair-per-thread.
// position == row index (cache_position is arange(S)).
// ---------------------------------------------------------------------------
__global__ __launch_bounds__(256)
void rope_k_kernel(const float* __restrict__ ckv, const float* __restrict__ cosp,
                   const float* __restrict__ sinp, float* __restrict__ krot)
{
  int idx = blockIdx.x * 256 + threadIdx.x;  // S_*32 threads
  int i = idx & 31;
  int s = idx >> 5;
  const float* x  = ckv + (size_t)s * (KVRANK_ + D_ROPE_) + KVRANK_;
  const float* c  = cosp + (size_t)s * D_ROPE_;
  const float* sn = sinp + (size_t)s * D_ROPE_;
  float x1 = x[i], x2 = x[i + 32];
  krot[(size_t)s * D_ROPE_ + i]      = x1 * c[i]      - x2 * sn[i];
  krot[(size_t)s * D_ROPE_ + i + 32] = x2 * c[i + 32] + x1 * sn[i + 32];
}

// ---------------------------------------------------------------------------
// In-place RoPE on q[:, h*192+128 : h*192+192]; pair-per-thread (race-free).
// ---------------------------------------------------------------------------
__global__ __launch_bounds__(256)
void rope_q_kernel(float* __restrict__ q, const float* __restrict__ cosp,
                   const float* __restrict__ sinp)
{
  int idx = blockIdx.x * 256 + threadIdx.x;  // S_*H_*32 threads
  int i = idx & 31;
  int h = (idx >> 5) & 15;
  int s = idx >> 9;
  float* x = q + (size_t)s * (H_ * D_QK_) + h * D_QK_ + D_NOPE_;
  const float* c  = cosp + (size_t)s * D_ROPE_;
  const float* sn = sinp + (size_t)s * D_ROPE_;
  float x1 = x[i], x2 = x[i + 32];
  x[i]      = x1 * c[i]      - x2 * sn[i];
  x[i + 32] = x2 * c[i + 32] + x1 * sn[i + 32];
}

// ---------------------------------------------------------------------------
// Kfull[kvh,s,0:128] = kv[s, kvh*256+d]; Kfull[kvh,s,128:192] = krot[s,:]
// ---------------------------------------------------------------------------
__global__ __launch_bounds__(256)
void assemble_k_kernel(const float* __restrict__ kv, const float* __restrict__ krot,
                       float* __restrict__ kfull)
{
  int idx = blockIdx.x * 256 + threadIdx.x;  // KVH_*S_*D_QK_
  int d    = idx % D_QK_;
  int rest = idx / D_QK_;
  int s    = rest & (S_ - 1);
  int kvh  = rest >> 11;
  float v = (d < D_NOPE_)
              ? kv[(size_t)s * (KVH_ * 256) + kvh * 256 + d]
              : krot[(size_t)s * D_ROPE_ + (d - D_NOPE_)];
  kfull[idx] = v;
}

// ---------------------------------------------------------------------------
// Flash attention, one wave per (16-query tile, head). Causal.
// K_j (16x192) and V_j (16x128) tiles are async-DMAed into double-buffered
// LDS one block ahead, so the 48 QK WMMAs + softmax + 32 PV WMMAs of block j
// overlap the DMA of block j+1. Row reductions via __shfl_xor align with the
// WMMA C/D layout (VGPR r <-> row r / r+8 per lane half). P is re-shaped
// D-layout -> A-layout through 1 KB of LDS.
// ---------------------------------------------------------------------------
__global__ __launch_bounds__(32)
void flash_attn_head(const float* __restrict__ Q,    // (S, H*D_QK)
                     const float* __restrict__ Kf,   // (KVH, S, D_QK)
                     const float* __restrict__ KVb,  // (S, KVH*256), V at +128
                     float* __restrict__ O)          // (S, H*D_V)
{
  __shared__ float sK[2][16 * 192];   // 12 KB per buffer
  __shared__ float sV[2][16 * 128];   //  8 KB per buffer
  __shared__ float pbuf[16 * 16];

  const int lane  = threadIdx.x;      // block = 32 (one wave)
  const int mrow  = lane & 15;
  const int half  = lane >> 4;
  const int khalf = half * 2;
  const int h    = blockIdx.y;
  const int qblk = blockIdx.x;        // 0..S/16-1
  const int q0   = qblk * 16;
  const int kvh  = h >> 2;            // GROUPS = 4
  const float scale = 0.07216878364870323f;  // 1/sqrt(192)
  const float NEG_INF = -__builtin_inff();

  const float* Kh = Kf + (size_t)kvh * S_ * D_QK_;
  const float* Vh = KVb + kvh * 256 + D_NOPE_;   // row stride KVH_*256

  // Async-stage K_j / V_j tiles: 768 + 512 float4, 24 + 16 b128 per lane.
  auto issue_kv = [&](int j, int buf) {
    const float* kbase = Kh + (size_t)(j * 16) * D_QK_;
#pragma unroll
    for (int i = 0; i < 24; ++i) {
      int f = lane + 32 * i;
      int r = f / 48, c4 = (f % 48) << 2;
      async_ld_b128(kbase + (size_t)r * D_QK_ + c4, &sK[buf][r * 192 + c4]);
    }
    const float* vbase = Vh + (size_t)(j * 16) * (KVH_ * 256);
#pragma unroll
    for (int i = 0; i < 16; ++i) {
      int f = lane + 32 * i;
      int r = f >> 5, c4 = (f & 31) << 2;
      async_ld_b128(vbase + (size_t)r * (KVH_ * 256) + c4, &sV[buf][r * 128 + c4]);
    }
  };

  // Preload Q fragments for all 48 K-steps (16x192 tile lives in registers).
  const float* qrow = Q + (size_t)(q0 + mrow) * (H_ * D_QK_) + h * D_QK_;
  v2f qf[48];
#pragma unroll
  for (int kk = 0; kk < 48; ++kk) {
    qf[kk].x = qrow[kk * 4 + khalf];
    qf[kk].y = qrow[kk * 4 + khalf + 1];
  }

  v8f acc[8] = {};
  float mrw[8], lrw[8];
#pragma unroll
  for (int r = 0; r < 8; ++r) { mrw[r] = NEG_INF; lrw[r] = 0.f; }

  issue_kv(0, 0);
  for (int j = 0; j <= qblk; ++j) {
    wait_async0();                      // buf j&1 resident
    if (j < qblk) issue_kv(j + 1, (j + 1) & 1);   // overlap with compute

    const float* Kb = &sK[j & 1][0];
    const float* Vb = &sV[j & 1][0];

    // ---- scores S = Q K_j^T (B-frag == row-major K rows, from LDS) ----
    v8f sc = {};
    const float* krow = Kb + mrow * 192 + khalf;
#pragma unroll
    for (int kk = 0; kk < 48; ++kk) {
      v2f b = *(const v2f*)(krow + kk * 4);
      sc = WMMA_F32(qf[kk], b, sc);
    }

    const bool diag = (j == qblk);
    float alpha[8];
#pragma unroll
    for (int r = 0; r < 8; ++r) {
      float v = sc[r] * scale;
      if (diag) {
        int row = r + (half << 3);
        if (mrow > row) v = -1.0e30f;   // key > query inside diagonal block
      }
      float red = v;                     // row max over the 16-lane half
      red = fmaxf(red, __shfl_xor(red, 1, 32));
      red = fmaxf(red, __shfl_xor(red, 2, 32));
      red = fmaxf(red, __shfl_xor(red, 4, 32));
      red = fmaxf(red, __shfl_xor(red, 8, 32));
      float newm = fmaxf(mrw[r], red);
      alpha[r] = __expf(mrw[r] - newm);
      mrw[r] = newm;
      float pe = __expf(v - newm);
      sc[r] = pe;
      float rs = pe;                     // row sum
      rs += __shfl_xor(rs, 1, 32);
      rs += __shfl_xor(rs, 2, 32);
      rs += __shfl_xor(rs, 4, 32);
      rs += __shfl_xor(rs, 8, 32);
      lrw[r] = lrw[r] * alpha[r] + rs;
    }

    // ---- P: D-layout -> A-layout via LDS ----
    __syncthreads();
#pragma unroll
    for (int r = 0; r < 8; ++r)
      pbuf[(r + (half << 3)) * 16 + mrow] = sc[r];
    __syncthreads();
    v2f pf[4];
#pragma unroll
    for (int kk = 0; kk < 4; ++kk) {
      pf[kk].x = pbuf[mrow * 16 + kk * 4 + khalf];
      pf[kk].y = pbuf[mrow * 16 + kk * 4 + khalf + 1];
    }

    // ---- O = O*alpha + P V_j (V from LDS) ----
#pragma unroll
    for (int t = 0; t < 8; ++t)
#pragma unroll
      for (int r = 0; r < 8; ++r) acc[t][r] *= alpha[r];

    const float* vb0 = Vb + khalf * 128 + mrow;
#pragma unroll
    for (int kk = 0; kk < 4; ++kk) {
      const float* v0 = vb0 + kk * 4 * 128;
#pragma unroll
      for (int t = 0; t < 8; ++t) {
        v2f b;
        b.x = v0[t * 16];
        b.y = v0[128 + t * 16];
        acc[t] = WMMA_F32(pf[kk], b, acc[t]);
      }
    }
  }

  // ---- epilogue: O / l ----
#pragma unroll
  for (int t = 0; t < 8; ++t)
#pragma unroll
    for (int r = 0; r < 8; ++r) {
      int row = q0 + r + (half << 3);
      O[(size_t)row * (H_ * D_V_) + h * D_V_ + t * 16 + mrow] = acc[t][r] / lrw[r];
    }
}

// ---------------------------------------------------------------------------
// Host-side pipeline.
// ---------------------------------------------------------------------------
extern "C" void kernel_launch(void* const* d_in, const int* in_sizes, int n_in,
                              void* d_out, int out_size, void* d_ws, size_t ws_size,
                              hipStream_t stream)
{
  (void)in_sizes; (void)n_in; (void)out_size; (void)ws_size;

  const float* hidden  = (const float*)d_in[0];
  const float* cosp    = (const float*)d_in[1];
  const float* sinp    = (const float*)d_in[2];
  const float* wq_a    = (const float*)d_in[3];
  const float* q_ln_w  = (const float*)d_in[4];
  const float* wq_b    = (const float*)d_in[5];
  const float* wkv_a   = (const float*)d_in[6];
  const float* kv_ln_w = (const float*)d_in[7];
  const float* wkv_b   = (const float*)d_in[8];
  const float* wo      = (const float*)d_in[9];
  float* out = (float*)d_out;

  // Workspace carve-up (floats). Total ~91 MB.
  float* ws   = (float*)d_ws;
  float* qa   = ws;                                      // S * QRANK
  float* qan  = qa   + (size_t)S_ * QRANK_;              // S * QRANK
  float* q    = qan  + (size_t)S_ * QRANK_;              // S * H*D_QK
  float* ckv  = q    + (size_t)S_ * H_ * D_QK_;          // S * 576
  float* ckvn = ckv  + (size_t)S_ * (KVRANK_ + D_ROPE_); // S * 512
  float* kv   = ckvn + (size_t)S_ * KVRANK_;             // S * 1024
  float* krot = kv   + (size_t)S_ * KVH_ * 256;          // S * 64
  float* kful = krot + (size_t)S_ * D_ROPE_;             // KVH * S * 192
  float* ao   = kful + (size_t)KVH_ * S_ * D_QK_;        // S * 2048

  // 1) q_a = hidden @ wq_a                      (2048 x 1536, K=2048)
  gemm_wmma_f32<<<dim3(QRANK_ / 64, S_ / 64), 128, 0, stream>>>(
      hidden, wq_a, qa, D_, D_, QRANK_, QRANK_);
  // 2) rmsnorm(q_a)
  rmsnorm_rows<<<S_, 256, 0, stream>>>(qa, q_ln_w, qan, QRANK_, QRANK_, QRANK_);
  // 3) q = q_a_n @ wq_b                         (2048 x 3072, K=1536)
  gemm_wmma_f32<<<dim3(H_ * D_QK_ / 64, S_ / 64), 128, 0, stream>>>(
      qan, wq_b, q, QRANK_, QRANK_, H_ * D_QK_, H_ * D_QK_);
  // 4) ckv = hidden @ wkv_a                     (2048 x 576, K=2048)
  gemm_wmma_f32<<<dim3((KVRANK_ + D_ROPE_) / 64, S_ / 64), 128, 0, stream>>>(
      hidden, wkv_a, ckv, D_, D_, KVRANK_ + D_ROPE_, KVRANK_ + D_ROPE_);
  // 5) rmsnorm(ckv[:, :512])
  rmsnorm_rows<<<S_, 256, 0, stream>>>(ckv, kv_ln_w, ckvn, KVRANK_,
                                       KVRANK_ + D_ROPE_, KVRANK_);
  // 6) kv = ckv_n @ wkv_b                       (2048 x 1024, K=512)
  gemm_wmma_f32<<<dim3(KVH_ * 256 / 64, S_ / 64), 128, 0, stream>>>(
      ckvn, wkv_b, kv, KVRANK_, KVRANK_, KVH_ * 256, KVH_ * 256);
  // 7) RoPE for shared k_rot
  rope_k_kernel<<<(S_ * 32) / 256, 256, 0, stream>>>(ckv, cosp, sinp, krot);
  // 8) RoPE on q (in place, pair-per-thread)
  rope_q_kernel<<<(S_ * H_ * 32) / 256, 256, 0, stream>>>(q, cosp, sinp);
  // 9) assemble K per kv-head: [k_pass | k_rot]
  assemble_k_kernel<<<(KVH_ * S_ * D_QK_) / 256, 256, 0, stream>>>(kv, krot, kful);
  // 10) flash attention (causal), 16 heads
  flash_attn_head<<<dim3(S_ / 16, H_), 32, 0, stream>>>(q, kful, kv, ao);
  // 11) out = attn_out @ wo                     (2048 x 2048, K=2048)
  gemm_wmma_f32<<<dim3(D_ / 64, S_ / 64), 128, 0, stream>>>(
      ao, wo, out, H_ * D_V_, H_ * D_V_, D_, D_);
}